// PLRNN_42399917146438
// MI455X (gfx1250) — compile-verified
//
#include <hip/hip_runtime.h>

#define DZ 64
#define DX 128
#define TT 100000
#define ZTILE 32   // 100000 % 32 == 0

typedef __attribute__((ext_vector_type(2))) float v2f;
typedef __attribute__((ext_vector_type(8))) float v8f;

// ---------------------------------------------------------------------------
// Kernel 1: sequential PLRNN recurrence. One workgroup, 64 threads (2 waves),
// thread d owns row d of W (in VGPRs) and state z[d]. relu(z) is shared via
// double-buffered LDS (broadcast reads). Z is staged in an LDS tile and
// dumped with contiguous float4 stores every ZTILE steps.
// ---------------------------------------------------------------------------
__launch_bounds__(64)
__global__ void plrnn_recur_kernel(const float* __restrict__ AW,
                                   const float* __restrict__ h,
                                   const float* __restrict__ Q0,
                                   const float* __restrict__ Q,
                                   const float* __restrict__ eps0,
                                   const float* __restrict__ eps,
                                   float* __restrict__ Zout) {
  const int d = threadIdx.x;                    // row 0..63

  __shared__ float zsh[2][DZ];                  // relu(z), double buffered
  __shared__ float ztile[DZ][ZTILE];            // staging for Z write-out

  // Load row d of W into registers; a = diag(AW), W diag zeroed.
  float Wrow[DZ];
  const float a = AW[d * DZ + d];
#pragma unroll
  for (int k = 0; k < DZ; ++k) Wrow[k] = AW[d * DZ + k];
  Wrow[d] = 0.0f;

  const float hd = h[d];
  const float qd = Q[d];

  float z = Q0[d] * eps0[d];                    // z(0)
  int buf = 0;
  zsh[buf][d] = fmaxf(z, 0.0f);
  __syncthreads();

  for (int tb = 0; tb < TT / ZTILE; ++tb) {
    for (int j = 0; j < ZTILE; ++j) {
      const int t = tb * ZTILE + j;
      float acc = fmaf(qd, eps[t * DZ + d], hd);
      acc = fmaf(a, z, acc);

      const float4* zr = (const float4*)&zsh[buf][0];   // broadcast LDS reads
#pragma unroll
      for (int k = 0; k < DZ / 4; ++k) {
        const float4 zv = zr[k];
        acc = fmaf(Wrow[4 * k + 0], zv.x, acc);
        acc = fmaf(Wrow[4 * k + 1], zv.y, acc);
        acc = fmaf(Wrow[4 * k + 2], zv.z, acc);
        acc = fmaf(Wrow[4 * k + 3], zv.w, acc);
      }

      z = acc;
      buf ^= 1;
      zsh[buf][d] = fmaxf(z, 0.0f);             // publish relu(z_{t+1})
      ztile[d][j] = z;                          // private staging row
      __syncthreads();                          // one barrier per step
    }
    // Dump: thread d writes its ZTILE contiguous elements of row d.
    float4* dst = (float4*)&Zout[(size_t)d * TT + (size_t)tb * ZTILE];
    const float4* src = (const float4*)&ztile[d][0];
#pragma unroll
    for (int j = 0; j < ZTILE / 4; ++j) dst[j] = src[j];
    // ztile row d is written/read only by thread d: no extra barrier needed.
  }
}

// ---------------------------------------------------------------------------
// Kernel 2: X = B @ Z + R * eta^T  via V_WMMA_F32_16X16X4_F32.
// Block = 256 threads (8 waves). Wave w computes the 16-row M-tile w for one
// 16-column t-slab (blockIdx.x). K=64 -> 16 WMMA ops per tile.
//
// f32 WMMA fragment layouts (ISA 7.12.2):
//   A (16x4):  lane%16 = M row,  K = vgpr_j + 2*(lane/16)
//   B (4x16):  lane%16 = N col,  K = vgpr_j + 2*(lane/16)
//   D (16x16): row = vgpr_v + 8*(lane/16), col = lane%16
// ---------------------------------------------------------------------------
__launch_bounds__(256)
__global__ void plrnn_emit_kernel(const float* __restrict__ Bw,
                                  const float* __restrict__ R,
                                  const float* __restrict__ Z,
                                  const float* __restrict__ eta,
                                  float* __restrict__ X) {
  const int wave = threadIdx.x >> 5;            // 0..7 -> M tile
  const int lane = threadIdx.x & 31;
  const int half = lane >> 4;                   // 0/1
  const int l    = lane & 15;

  const int m0 = wave * 16;
  const int t0 = blockIdx.x * 16;

  v8f acc = {};

#pragma unroll
  for (int k0 = 0; k0 < DZ; k0 += 4) {
    const int ka = k0 + 2 * half;
    v2f av, bv;
    av.x = Bw[(m0 + l) * DZ + ka + 0];          // A: rows of B-weights
    av.y = Bw[(m0 + l) * DZ + ka + 1];
    bv.x = Z[(size_t)(ka + 0) * TT + t0 + l];   // B: rows of Z (64B/half-wave)
    bv.y = Z[(size_t)(ka + 1) * TT + t0 + l];
    acc = __builtin_amdgcn_wmma_f32_16x16x4_f32(
        /*neg_a=*/false, av, /*neg_b=*/false, bv,
        /*c_mod=*/(short)0, acc, /*reuse_a=*/false, /*reuse_b=*/false);
  }

  // Epilogue: add R[m] * eta[t][m] and store X[m][t].
  const int t = t0 + l;
#pragma unroll
  for (int v = 0; v < 8; ++v) {
    const int m = m0 + v + 8 * half;
    X[(size_t)m * TT + t] = acc[v] + R[m] * eta[(size_t)t * DX + m];
  }
}

// ---------------------------------------------------------------------------
// Launch. Inputs (setup_inputs order):
//   0 AW(64x64) 1 h(64) 2 B(128x64) 3 Q0(64) 4 Q(64) 5 R(128)
//   6 eps0(64)  7 eps(T x 64)  8 eta(T x 128)
// d_out = [X (128*T) | Z (64*T)] float32.
// ---------------------------------------------------------------------------
extern "C" void kernel_launch(void* const* d_in, const int* in_sizes, int n_in,
                              void* d_out, int out_size, void* d_ws, size_t ws_size,
                              hipStream_t stream) {
  const float* AW   = (const float*)d_in[0];
  const float* h    = (const float*)d_in[1];
  const float* Bw   = (const float*)d_in[2];
  const float* Q0   = (const float*)d_in[3];
  const float* Q    = (const float*)d_in[4];
  const float* R    = (const float*)d_in[5];
  const float* eps0 = (const float*)d_in[6];
  const float* eps  = (const float*)d_in[7];
  const float* eta  = (const float*)d_in[8];

  float* Xo = (float*)d_out;                    // (DX, T)
  float* Zo = Xo + (size_t)DX * TT;             // (DZ, T)

  plrnn_recur_kernel<<<1, 64, 0, stream>>>(AW, h, Q0, Q, eps0, eps, Zo);
  plrnn_emit_kernel<<<TT / 16, 256, 0, stream>>>(Bw, R, Zo, eta, Xo);
}